// Test_Model_86105504350704
// MI455X (gfx1250) — compile-verified
//
#include <hip/hip_runtime.h>
#include <math.h>

// ---------------- problem constants -------------------------------------
#define MAXL   15
#define NL     16          // MAXL+1
#define NPAIR  256         // NL*NL
#define TAU    4
#define BATCH  32
// per-batch flattened output length (per complex component):
//   16 * (sum_l (2l+1))^2 = 16 * 256^2
#define NFLAT  1048576
// dense CG storage: sum over pairs of ((2l1+1)(2l2+1))^2 = 5456^2 floats
#define CG_TOTAL 29767936L
// total GEMM rows across all pairs: (sum (2l+1))^2
#define ROWS_TOTAL 65536
#define MSTRIPS 2          // 16-row strips per wave (4 WMMAs per B build)

typedef __attribute__((ext_vector_type(2))) float v2f;
typedef __attribute__((ext_vector_type(8))) float v8f;

__device__ __forceinline__ int AL(int l)   { return 2*l + 1; }
__device__ __forceinline__ int iabs(int x) { return x < 0 ? -x : x; }

// ---------------- CG coefficient math (Racah, log space) ----------------
__device__ float cg_val(const double* __restrict__ lf,
                        int l1, int l2, int l, int m1, int m2) {
  int m = m1 + m2;
  double logpre = 0.5 * (lf[l1+l2-l] + lf[l1-l2+l] + lf[-l1+l2+l]
                         - lf[l1+l2+l+1]);
  double lognum = 0.5 * (lf[l+m] + lf[l-m] + lf[l1-m1] + lf[l1+m1]
                         + lf[l2-m2] + lf[l2+m2]);
  double s = 0.0;
  int kmax = l1 + l2 - l;
  for (int k = 0; k <= kmax; ++k) {
    int a1 = l1+l2-l-k, a2 = l1-m1-k, a3 = l2+m2-k;
    int a4 = l-l2+m1+k, a5 = l-l1-m2+k;
    if (a1 >= 0 && a2 >= 0 && a3 >= 0 && a4 >= 0 && a5 >= 0) {
      double logden = lf[k]+lf[a1]+lf[a2]+lf[a3]+lf[a4]+lf[a5];
      s += ((k & 1) ? -1.0 : 1.0) * exp(-logden);
    }
  }
  return (float)(sqrt(2.0*(double)l + 1.0) * exp(logpre + lognum) * s);
}

// ---------------- output-layout bookkeeping -----------------------------
__device__ int count_pairs_for_l(int l) {
  int c = 0;
  for (int q = 0; q < NPAIR; ++q) {
    int l1 = q >> 4, l2 = q & 15;
    if (l >= iabs(l1-l2) && l <= l1+l2) ++c;
  }
  return c;
}
__device__ int pairidx_for_l(int p, int l) {
  int c = 0;
  for (int q = 0; q < p; ++q) {
    int l1 = q >> 4, l2 = q & 15;
    if (l >= iabs(l1-l2) && l <= l1+l2) ++c;
  }
  return c;
}
__device__ int offset_for_l(int l) {
  long o = 0;
  for (int ll = 0; ll < l; ++ll)
    o += (long)AL(ll) * 16 * count_pairs_for_l(ll);
  return (int)o;
}

// ---------------- kernel 1: row -> output-position map ------------------
__global__ void rowbase_kernel(int* __restrict__ rowbase) {
  int p  = blockIdx.x;
  int l1 = p >> 4, l2 = p & 15;
  int K  = AL(l1) * AL(l2);
  int lmin = iabs(l1-l2), lmax = l1+l2;
  int roff = 0;
  for (int q = 0; q < p; ++q) roff += AL(q >> 4) * AL(q & 15);

  __shared__ int s_off[32], s_P[32], s_pidx[32];
  int nl = lmax - lmin + 1;
  for (int t = threadIdx.x; t < nl; t += blockDim.x) {
    int l = lmin + t;
    s_off[t]  = offset_for_l(l);
    s_P[t]    = count_pairs_for_l(l);
    s_pidx[t] = pairidx_for_l(p, l);
  }
  __syncthreads();

  int lmin2 = lmin * lmin;
  for (int r = threadIdx.x; r < K; r += blockDim.x) {
    int v = r + lmin2;
    int l = (int)floorf(sqrtf((float)v));
    while ((l+1)*(l+1) <= v) ++l;
    while (l*l > v) --l;
    int k = v - l*l;
    int t = l - lmin;
    rowbase[roff + r] = s_off[t] + k * 16 * s_P[t] + s_pidx[t] * 16;
  }
}

// ---------------- kernel 2: dense CG matrices + (i,j) table -------------
// Cmat[row=(l,k), col=(i*a2+j)]; only k = (m1+m2)+l is nonzero per (i,j,l).
// ijtab[roff + kr] = (i<<8)|j   with kr = i*a2+j (B k-row decode table).
__global__ void cg_setup_kernel(float* __restrict__ ws,
                                int* __restrict__ ijtab) {
  __shared__ double s_lf[64];
  if (threadIdx.x == 0) {
    double s = 0.0;
    s_lf[0] = 0.0;
    for (int t = 1; t < 64; ++t) { s += log((double)t); s_lf[t] = s; }
  }
  __syncthreads();

  int p  = blockIdx.x;
  int l1 = p >> 4, l2 = p & 15;
  int a1 = AL(l1), a2 = AL(l2);
  int K  = a1 * a2;
  long coff = 0; int roff = 0;
  for (int q = 0; q < p; ++q) {
    long Kq = (long)AL(q >> 4) * AL(q & 15);
    coff += Kq * Kq; roff += (int)Kq;
  }
  float* C = ws + coff;
  for (long t = threadIdx.x; t < (long)K*K; t += blockDim.x) C[t] = 0.f;
  __syncthreads();

  int lmin = iabs(l1-l2), lmax = l1+l2;
  for (int idx = threadIdx.x; idx < K; idx += blockDim.x) {
    int i = idx / a2, j = idx % a2;
    ijtab[roff + idx] = (i << 8) | j;
    int m1 = i - l1, m2 = j - l2, m = m1 + m2;
    int lo = max(lmin, iabs(m));
    for (int l = lo; l <= lmax; ++l) {
      float v  = cg_val(s_lf, l1, l2, l, m1, m2);
      int row  = l*l - lmin*lmin + (m + l);
      C[(long)row * K + idx] = v;
    }
  }
}

// ---------------- kernel 3: WMMA GEMM + scatter epilogue ----------------
// Per pair: out[(l,k),(b,m,n)] = Cmat (KxK) x P (Kx512); re & im share A.
// Block = 4 waves; wave w owns batch b = blockIdx.y*4+w (16 cols = m*4+n)
// and MSTRIPS=2 16-row A strips -> 4 accumulators, 4 WMMAs per B build.
struct XPtrs { const float* x[NL]; };

__global__ void __launch_bounds__(128)
cg_gemm_kernel(XPtrs xp, const float* __restrict__ ws,
               const int* __restrict__ rowbase, const int* __restrict__ ijtab,
               float* __restrict__ out) {
  // decode (pair p, 32-row m-tile) from blockIdx.x
  int bx = blockIdx.x;
  int p = 0, mtile = 0, roff = 0, mt_cum = 0;
  long coff = 0;
  for (int q = 0; q < NPAIR; ++q) {
    int Kq = AL(q >> 4) * AL(q & 15);
    int mt = (Kq + 16*MSTRIPS - 1) / (16*MSTRIPS);
    if (bx < mt_cum + mt) { p = q; mtile = bx - mt_cum; break; }
    coff += (long)Kq * Kq;
    roff += Kq;
    mt_cum += mt;
  }
  int l1 = p >> 4, l2 = p & 15;
  int a1 = AL(l1), a2 = AL(l2), K = a1 * a2;
  const float* Cm  = ws + coff;
  const int*   tab = ijtab + roff;

  // stage the (tiny) inputs for this pair into LDS: layout == global layout
  __shared__ float sx1[BATCH * 31 * TAU * 2];
  __shared__ float sx2[BATCH * 31 * TAU * 2];
  {
    const float* x1g = xp.x[l1];
    const float* x2g = xp.x[l2];
    int n1 = BATCH * a1 * TAU * 2, n2 = BATCH * a2 * TAU * 2;
    for (int t = threadIdx.x; t < n1; t += blockDim.x) sx1[t] = x1g[t];
    for (int t = threadIdx.x; t < n2; t += blockDim.x) sx2[t] = x2g[t];
  }
  __syncthreads();

  int wave = threadIdx.x >> 5;
  int lane = threadIdx.x & 31;
  int b    = blockIdx.y * 4 + wave;     // batch (N-tile) for this wave
  int hi   = lane >> 4;                 // lane group (K split for A/B)
  int l15  = lane & 15;
  int m    = l15 >> 2, n = l15 & 3;     // channel indices for this column

  // A row pointers for the two strips (clamped addresses; an A row only
  // contributes to the same D row, and the epilogue skips rows >= K)
  int mrow0 = mtile * (16*MSTRIPS) + l15;
  int mrow1 = mrow0 + 16;
  const float* Ar0 = Cm + (size_t)min(mrow0, K-1) * K;
  const float* Ar1 = Cm + (size_t)min(mrow1, K-1) * K;

  // per-lane LDS row bases: + i*(TAU*2) / + j*(TAU*2)
  const float* sx1b = sx1 + (b * a1 * TAU + m) * 2;
  const float* sx2b = sx2 + (b * a2 * TAU + n) * 2;

  v8f accre[MSTRIPS]; v8f accim[MSTRIPS];
  accre[0] = (v8f){}; accre[1] = (v8f){};
  accim[0] = (v8f){}; accim[1] = (v8f){};

  int Kmain = K & ~3;
  #pragma unroll 2
  for (int kk = 0; kk < Kmain; kk += 4) {
    int kc = kk + 2*hi;                 // this lane's two K columns
    v2f a0, a1v;
    a0.x  = Ar0[kc]; a0.y  = Ar0[kc+1];
    a1v.x = Ar1[kc]; a1v.y = Ar1[kc+1];

    int t0 = tab[kc], t1 = tab[kc+1];
    const v2f X0 = *(const v2f*)(sx1b + (t0 >> 8)  * (TAU*2));
    const v2f Y0 = *(const v2f*)(sx2b + (t0 & 255) * (TAU*2));
    const v2f X1 = *(const v2f*)(sx1b + (t1 >> 8)  * (TAU*2));
    const v2f Y1 = *(const v2f*)(sx2b + (t1 & 255) * (TAU*2));

    v2f pre, pim;
    pre.x = X0.x*Y0.x - X0.y*Y0.y;  pim.x = X0.x*Y0.y + X0.y*Y0.x;
    pre.y = X1.x*Y1.x - X1.y*Y1.y;  pim.y = X1.x*Y1.y + X1.y*Y1.x;

    accre[0] = __builtin_amdgcn_wmma_f32_16x16x4_f32(
        false, a0,  false, pre, (short)0, accre[0], false, false);
    accim[0] = __builtin_amdgcn_wmma_f32_16x16x4_f32(
        false, a0,  false, pim, (short)0, accim[0], false, false);
    accre[1] = __builtin_amdgcn_wmma_f32_16x16x4_f32(
        false, a1v, false, pre, (short)0, accre[1], false, false);
    accim[1] = __builtin_amdgcn_wmma_f32_16x16x4_f32(
        false, a1v, false, pim, (short)0, accim[1], false, false);
  }

  if (Kmain < K) {                      // guarded tail (K % 4 != 0)
    int kc  = Kmain + 2*hi;
    int kc0 = min(kc, K-1), kc1 = min(kc+1, K-1);
    bool ok0 = kc < K, ok1 = kc+1 < K;
    v2f a0, a1v;
    a0.x  = ok0 ? Ar0[kc0] : 0.f;  a0.y  = ok1 ? Ar0[kc1] : 0.f;
    a1v.x = ok0 ? Ar1[kc0] : 0.f;  a1v.y = ok1 ? Ar1[kc1] : 0.f;

    int t0 = tab[kc0], t1 = tab[kc1];
    const v2f X0 = *(const v2f*)(sx1b + (t0 >> 8)  * (TAU*2));
    const v2f Y0 = *(const v2f*)(sx2b + (t0 & 255) * (TAU*2));
    const v2f X1 = *(const v2f*)(sx1b + (t1 >> 8)  * (TAU*2));
    const v2f Y1 = *(const v2f*)(sx2b + (t1 & 255) * (TAU*2));

    v2f pre, pim;
    pre.x = ok0 ? (X0.x*Y0.x - X0.y*Y0.y) : 0.f;
    pim.x = ok0 ? (X0.x*Y0.y + X0.y*Y0.x) : 0.f;
    pre.y = ok1 ? (X1.x*Y1.x - X1.y*Y1.y) : 0.f;
    pim.y = ok1 ? (X1.x*Y1.y + X1.y*Y1.x) : 0.f;

    accre[0] = __builtin_amdgcn_wmma_f32_16x16x4_f32(
        false, a0,  false, pre, (short)0, accre[0], false, false);
    accim[0] = __builtin_amdgcn_wmma_f32_16x16x4_f32(
        false, a0,  false, pim, (short)0, accim[0], false, false);
    accre[1] = __builtin_amdgcn_wmma_f32_16x16x4_f32(
        false, a1v, false, pre, (short)0, accre[1], false, false);
    accim[1] = __builtin_amdgcn_wmma_f32_16x16x4_f32(
        false, a1v, false, pim, (short)0, accim[1], false, false);
  }

  // epilogue: D layout VGPR v -> M = v + 8*hi, N = lane&15
  // stream {re,im} out with a non-temporal hint (keep CG data in L2)
  #pragma unroll
  for (int s = 0; s < MSTRIPS; ++s) {
    #pragma unroll
    for (int v = 0; v < 8; ++v) {
      int r = mtile * (16*MSTRIPS) + s*16 + v + hi*8;
      if (r < K) {
        int pos = rowbase[roff + r] + l15;
        v2f val;
        val.x = accre[s][v];
        val.y = accim[s][v];
        __builtin_nontemporal_store(val, (v2f*)out + (size_t)b * NFLAT + pos);
      }
    }
  }
}

// ---------------- host launcher -----------------------------------------
extern "C" void kernel_launch(void* const* d_in, const int* in_sizes, int n_in,
                              void* d_out, int out_size, void* d_ws, size_t ws_size,
                              hipStream_t stream) {
  (void)in_sizes; (void)n_in; (void)out_size; (void)ws_size;
  float* ws      = (float*)d_ws;
  int*   rowbase = (int*)(ws + CG_TOTAL);
  int*   ijtab   = rowbase + ROWS_TOTAL;

  XPtrs xp;
  for (int l = 0; l < NL; ++l) xp.x[l] = (const float*)d_in[l];

  hipLaunchKernelGGL(rowbase_kernel,  dim3(NPAIR), dim3(256), 0, stream, rowbase);
  hipLaunchKernelGGL(cg_setup_kernel, dim3(NPAIR), dim3(256), 0, stream, ws, ijtab);

  int total_mt = 0;
  for (int q = 0; q < NPAIR; ++q) {
    int K = (2*(q >> 4) + 1) * (2*(q & 15) + 1);
    total_mt += (K + 16*MSTRIPS - 1) / (16*MSTRIPS);
  }
  hipLaunchKernelGGL(cg_gemm_kernel, dim3(total_mt, 8), dim3(128), 0, stream,
                     xp, (const float*)ws, (const int*)rowbase,
                     (const int*)ijtab, (float*)d_out);
}